// TGN_25718264168724
// MI455X (gfx1250) — compile-verified
//
#include <hip/hip_runtime.h>
#include <math.h>

typedef __attribute__((ext_vector_type(2))) float v2f;
typedef __attribute__((ext_vector_type(8))) float v8f;

#define NN_NODES 200000
#define DD       256
#define DQn      512
#define DKn      768
#define NSn      4096
#define BQn      8192     // 2 * N_SAMPLES
#define KNn      20
#define MMSG     8192
#define CDIM     64

// ---------------------------------------------------------------------------
// Generic f32 WMMA GEMM: C[M,N] = op(A[M,K] (+row gather)) @ B[K,N] (+bias,relu)
// Wave32: each wave owns a 32x32 tile (2 M-frags x 2 N-frags, 4 accumulators,
// 4 x v_wmma_f32_16x16x4_f32 per K-chunk); 8 waves arranged 4x2 -> 128x64
// block tile. Requires M % 128 == 0, N % 64 == 0, K % 8 == 0.
// TRANSB / GATHER are compile-time so the K-loop is branch-free.
// ---------------------------------------------------------------------------
template <bool TRANSB, bool GATHER>
__global__ __launch_bounds__(256) void gemm_wmma_f32(
    const float* __restrict__ A, int lda, int aoff, const int* __restrict__ aidx,
    const float* __restrict__ Bw, int ldb, int boff,
    const float* __restrict__ bias,
    float* __restrict__ C, int ldc, int coff,
    int K, int do_relu)
{
    const int lane  = threadIdx.x & 31;
    const int wid   = threadIdx.x >> 5;
    const int waveM = wid >> 1;                 // 0..3
    const int waveN = wid & 1;                  // 0..1
    const int tileM = blockIdx.y * 128 + waveM * 32;
    const int tileN = blockIdx.x * 64  + waveN * 32;

    const int ncol  = lane & 15;
    const int khalf = (lane >> 4) << 1;         // 0 for lanes 0-15, 2 for 16-31

    // A row pointers for the two 16-row M-fragments
    const int  mr0 = tileM + ncol;
    const int  mr1 = mr0 + 16;
    const long ar0 = GATHER ? (long)aidx[mr0] : (long)mr0;
    const long ar1 = GATHER ? (long)aidx[mr1] : (long)mr1;
    const float* aptr0 = A + ar0 * (long)lda + aoff + khalf;
    const float* aptr1 = A + ar1 * (long)lda + aoff + khalf;

    // B base pointers for the two 16-col N-fragments (transB case)
    const float* bt0 = Bw + (long)(tileN + ncol)      * ldb + boff + khalf;
    const float* bt1 = Bw + (long)(tileN + 16 + ncol) * ldb + boff + khalf;

    v8f acc00 = {}, acc01 = {}, acc10 = {}, acc11 = {};

    for (int kk = 0; kk < K; kk += 8) {
        __builtin_prefetch(aptr0 + kk + 128, 0, 0);   // stream-ahead on A rows
        __builtin_prefetch(aptr1 + kk + 128, 0, 0);
        #pragma unroll
        for (int u = 0; u < 8; u += 4) {
            const int k4 = kk + u;
            v2f a0, a1, b0, b1;
            { const float* p = aptr0 + k4; a0.x = p[0]; a0.y = p[1]; }
            { const float* p = aptr1 + k4; a1.x = p[0]; a1.y = p[1]; }
            if (TRANSB) {
                const float* p0 = bt0 + k4; b0.x = p0[0]; b0.y = p0[1];
                const float* p1 = bt1 + k4; b1.x = p1[0]; b1.y = p1[1];
            } else {
                const long r0 = (long)(k4 + khalf) * ldb + boff;
                const long r1 = r0 + ldb;
                b0.x = Bw[r0 + tileN + ncol];       b0.y = Bw[r1 + tileN + ncol];
                b1.x = Bw[r0 + tileN + 16 + ncol];  b1.y = Bw[r1 + tileN + 16 + ncol];
            }
            acc00 = __builtin_amdgcn_wmma_f32_16x16x4_f32(false, a0, false, b0, (short)0, acc00, false, false);
            acc01 = __builtin_amdgcn_wmma_f32_16x16x4_f32(false, a0, false, b1, (short)0, acc01, false, false);
            acc10 = __builtin_amdgcn_wmma_f32_16x16x4_f32(false, a1, false, b0, (short)0, acc10, false, false);
            acc11 = __builtin_amdgcn_wmma_f32_16x16x4_f32(false, a1, false, b1, (short)0, acc11, false, false);
        }
    }

    // C 16x16 frag: VGPR i holds row i (lanes 0-15) / row 8+i (lanes 16-31)
    const int rbase = tileM + ((lane >> 4) << 3);
    const int c0    = tileN + ncol;
    const int c1    = c0 + 16;
    float bias0 = 0.f, bias1 = 0.f;
    if (bias) { bias0 = bias[coff + c0]; bias1 = bias[coff + c1]; }
    #pragma unroll
    for (int i = 0; i < 8; ++i) {
        const int row0 = rbase + i;
        const int row1 = row0 + 16;
        float v00 = acc00[i] + bias0, v01 = acc01[i] + bias1;
        float v10 = acc10[i] + bias0, v11 = acc11[i] + bias1;
        if (do_relu) {
            v00 = fmaxf(v00, 0.f); v01 = fmaxf(v01, 0.f);
            v10 = fmaxf(v10, 0.f); v11 = fmaxf(v11, 0.f);
        }
        C[(long)row0 * ldc + coff + c0] = v00;
        C[(long)row0 * ldc + coff + c1] = v01;
        C[(long)row1 * ldc + coff + c0] = v10;
        C[(long)row1 * ldc + coff + c1] = v11;
    }
}

// ---------------------------------------------------------------------------
// Small elementwise / gather kernels
// ---------------------------------------------------------------------------
__global__ void k_init_updidx(int* __restrict__ u, int n) {
    int i = blockIdx.x * blockDim.x + threadIdx.x;
    if (i < n) u[i] = -1;
}

__global__ void k_scatter_updidx(int* __restrict__ u, const int* __restrict__ ids, int m) {
    int i = blockIdx.x * blockDim.x + threadIdx.x;
    if (i < m) u[ids[i]] = i;
}

__global__ void k_gru_combine(const float* __restrict__ GI, const float* __restrict__ GH,
                              const float* __restrict__ memory, const int* __restrict__ ids,
                              float* __restrict__ newmem) {
    long i = (long)blockIdx.x * blockDim.x + threadIdx.x;
    if (i >= (long)MMSG * DD) return;
    int j = (int)(i / DD), d = (int)(i % DD);
    const float* gi = GI + (long)j * (3 * DD);
    const float* gh = GH + (long)j * (3 * DD);
    float r = 1.0f / (1.0f + expf(-(gi[d]          + gh[d])));
    float z = 1.0f / (1.0f + expf(-(gi[DD + d]     + gh[DD + d])));
    float g = tanhf(gi[2 * DD + d] + r * gh[2 * DD + d]);
    float h = memory[(long)ids[j] * DD + d];
    newmem[(long)j * DD + d] = (1.0f - z) * g + z * h;
}

__global__ void k_build_query(const float* __restrict__ nodef, const float* __restrict__ memory,
                              const float* __restrict__ newmem, const int* __restrict__ updidx,
                              const int* __restrict__ src, const int* __restrict__ dst,
                              const float* __restrict__ time_b,
                              float* __restrict__ QF, float* __restrict__ SF) {
    long i = (long)blockIdx.x * blockDim.x + threadIdx.x;
    if (i >= (long)BQn * DD) return;
    int b = (int)(i / DD), d = (int)(i % DD);
    int node = (b < NSn) ? src[b] : dst[b - NSn];
    int j = updidx[node];
    float mv = (j >= 0) ? newmem[(long)j * DD + d] : memory[(long)node * DD + d];
    float sf = mv + nodef[(long)node * DD + d];
    SF[(long)b * DD + d]        = sf;
    QF[(long)b * DQn + d]       = sf;
    QF[(long)b * DQn + DD + d]  = cosf(time_b[d]);   // tenc(0) = cos(0*w + b)
}

__global__ void k_score_bias(const float* __restrict__ Q, const float* __restrict__ bk,
                             float* __restrict__ sb) {
    int i = blockIdx.x * blockDim.x + threadIdx.x;
    if (i >= BQn * 2) return;
    int b = i >> 1, h = i & 1;
    const float* q  = Q + (long)b * DQn + h * 256;
    const float* bb = bk + h * 256;
    float s = 0.0f;
    for (int d = 0; d < 256; ++d) s += q[d] * bb[d];
    sb[i] = s;
}

// Fused per-query attention: build 20 key rows (gathers + time-enc) into LDS,
// score against T[b,h,:] (= Wk_h^T q_h), softmax over K=20, and emit
// U[b,h,:] = sum_k attn * key_row  (value projection folded out of the loop).
__global__ __launch_bounds__(256) void k_attn_fused(
    const float* __restrict__ nodef, const float* __restrict__ memory,
    const float* __restrict__ newmem, const int* __restrict__ updidx,
    const float* __restrict__ edgef, const float* __restrict__ time_w,
    const float* __restrict__ time_b, const float* __restrict__ edge_times,
    const float* __restrict__ neigh_ts, const int* __restrict__ neigh_ids,
    const int* __restrict__ neigh_eidx,
    const float* __restrict__ T, const float* __restrict__ sb,
    float* __restrict__ U)
{
    __shared__ float sk[KNn * DKn];   // 20 x 768 key rows (61440 B < 64KB)
    __shared__ float ss[KNn * 2];
    __shared__ float sa[KNn * 2];

    const int b   = blockIdx.x;
    const int tid = threadIdx.x;
    const float ts = edge_times[(b < NSn) ? b : (b - NSn)];

    for (int idx = tid; idx < KNn * DKn; idx += 256) {
        int k = idx / DKn, d = idx - k * DKn;
        float v;
        if (d < 256) {
            int node = neigh_ids[(long)b * KNn + k];
            int j = updidx[node];
            float mv = (j >= 0) ? newmem[(long)j * DD + d] : memory[(long)node * DD + d];
            v = mv + nodef[(long)node * DD + d];
        } else if (d < 512) {
            int e = neigh_eidx[(long)b * KNn + k];
            v = edgef[(long)e * DD + (d - 256)];
        } else {
            float dt = ts - neigh_ts[(long)b * KNn + k];
            v = cosf(dt * time_w[d - 512] + time_b[d - 512]);
        }
        sk[idx] = v;
    }
    __syncthreads();

    const int lane = tid & 31, wid = tid >> 5;
    const float* Tb = T + (long)b * 2 * DKn;
    for (int p = wid; p < KNn * 2; p += 8) {
        int k = p >> 1, h = p & 1;
        const float* kr = sk + k * DKn;
        const float* tr = Tb + h * DKn;
        float s = 0.0f;
        for (int d = lane; d < DKn; d += 32) s += kr[d] * tr[d];
        for (int off = 16; off > 0; off >>= 1) s += __shfl_xor(s, off, 32);
        if (lane == 0) ss[p] = (s + sb[b * 2 + h]) * 0.0625f;   // 1/sqrt(dh=256)
    }
    __syncthreads();

    if (tid < 2) {
        int h = tid;
        float mx = -1e30f;
        for (int k = 0; k < KNn; ++k) mx = fmaxf(mx, ss[k * 2 + h]);
        float sum = 0.0f;
        for (int k = 0; k < KNn; ++k) { float e = expf(ss[k * 2 + h] - mx); sa[k * 2 + h] = e; sum += e; }
        float inv = 1.0f / sum;
        for (int k = 0; k < KNn; ++k) sa[k * 2 + h] *= inv;
    }
    __syncthreads();

    for (int d = tid; d < DKn; d += 256) {
        float u0 = 0.0f, u1 = 0.0f;
        for (int k = 0; k < KNn; ++k) {
            float kv = sk[k * DKn + d];
            u0 += sa[k * 2 + 0] * kv;
            u1 += sa[k * 2 + 1] * kv;
        }
        U[((long)b * 2 + 0) * DKn + d] = u0;
        U[((long)b * 2 + 1) * DKn + d] = u1;
    }
}

__global__ void k_build_cat(const float* __restrict__ O, const float* __restrict__ SF,
                            float* __restrict__ CAT) {
    long i = (long)blockIdx.x * blockDim.x + threadIdx.x;
    if (i >= (long)BQn * DKn) return;
    int b = (int)(i / DKn), d = (int)(i % DKn);
    CAT[i] = (d < DQn) ? O[(long)b * DQn + d] : SF[(long)b * DD + (d - DQn)];
}

__global__ void k_softmax_rows(const float* __restrict__ P, float* __restrict__ out) {
    int b = blockIdx.x * blockDim.x + threadIdx.x;
    if (b >= BQn) return;
    const float* p = P + (long)b * CDIM;
    float mx = -1e30f;
    for (int c = 0; c < CDIM; ++c) mx = fmaxf(mx, p[c]);
    float sum = 0.0f;
    for (int c = 0; c < CDIM; ++c) sum += expf(p[c] - mx);
    float inv = 1.0f / sum;
    for (int c = 0; c < CDIM; ++c) out[(long)b * CDIM + c] = expf(p[c] - mx) * inv;
}

// ---------------------------------------------------------------------------
// Host side
// ---------------------------------------------------------------------------
static inline void gemm(hipStream_t s,
                        const float* A, int lda, int aoff, const int* aidx,
                        const float* Bw, int ldb, int boff, int transB,
                        const float* bias,
                        float* C, int ldc, int coff,
                        int M, int N, int K, int relu) {
    dim3 grid(N / 64, M / 128);
    if (aidx)
        gemm_wmma_f32<false, true><<<grid, 256, 0, s>>>(A, lda, aoff, aidx, Bw, ldb, boff,
                                                        bias, C, ldc, coff, K, relu);
    else if (transB)
        gemm_wmma_f32<true, false><<<grid, 256, 0, s>>>(A, lda, aoff, aidx, Bw, ldb, boff,
                                                        bias, C, ldc, coff, K, relu);
    else
        gemm_wmma_f32<false, false><<<grid, 256, 0, s>>>(A, lda, aoff, aidx, Bw, ldb, boff,
                                                         bias, C, ldc, coff, K, relu);
}

extern "C" void kernel_launch(void* const* d_in, const int* in_sizes, int n_in,
                              void* d_out, int out_size, void* d_ws, size_t ws_size,
                              hipStream_t stream) {
    (void)in_sizes; (void)n_in; (void)out_size; (void)ws_size;

    const float* node_features = (const float*)d_in[0];
    const float* edge_features = (const float*)d_in[1];
    const float* memory        = (const float*)d_in[2];
    const float* time_w        = (const float*)d_in[3];
    const float* time_b        = (const float*)d_in[4];
    const float* msg_w1        = (const float*)d_in[5];
    const float* msg_b1        = (const float*)d_in[6];
    const float* msg_w2        = (const float*)d_in[7];
    const float* msg_b2        = (const float*)d_in[8];
    const float* gru_wi        = (const float*)d_in[9];
    const float* gru_wh        = (const float*)d_in[10];
    const float* gru_bi        = (const float*)d_in[11];
    const float* gru_bh        = (const float*)d_in[12];
    const float* wq            = (const float*)d_in[13];
    const float* bq            = (const float*)d_in[14];
    const float* wk            = (const float*)d_in[15];
    const float* bk            = (const float*)d_in[16];
    const float* wv            = (const float*)d_in[17];
    const float* bv            = (const float*)d_in[18];
    const float* wo            = (const float*)d_in[19];
    const float* bo            = (const float*)d_in[20];
    const float* mrg_w1        = (const float*)d_in[21];
    const float* mrg_b1        = (const float*)d_in[22];
    const float* mrg_w2        = (const float*)d_in[23];
    const float* mrg_b2        = (const float*)d_in[24];
    const float* prj_w1        = (const float*)d_in[25];
    const float* prj_b1        = (const float*)d_in[26];
    const float* prj_w2        = (const float*)d_in[27];
    const float* prj_b2        = (const float*)d_in[28];
    const float* edge_times    = (const float*)d_in[29];
    const float* neigh_ts      = (const float*)d_in[30];
    const float* msg_raw       = (const float*)d_in[31];
    const int*   src_nodes     = (const int*)d_in[32];
    const int*   dst_nodes     = (const int*)d_in[33];
    const int*   neigh_ids     = (const int*)d_in[34];
    const int*   neigh_eidx    = (const int*)d_in[35];
    const int*   msg_node_ids  = (const int*)d_in[36];

    // workspace carve-out (256B aligned)
    char*  ws  = (char*)d_ws;
    size_t off = 0;
    auto carve = [&](size_t bytes) -> void* {
        void* p = ws + off;
        off += (bytes + 255) & ~(size_t)255;
        return p;
    };
    int*   upd_idx = (int*)  carve((size_t)NN_NODES * 4);
    float* H1      = (float*)carve((size_t)MMSG * 512 * 4);
    float* Mm      = (float*)carve((size_t)MMSG * 256 * 4);
    float* GI      = (float*)carve((size_t)MMSG * 768 * 4);
    float* GH      = (float*)carve((size_t)MMSG * 768 * 4);
    float* newmem  = (float*)carve((size_t)MMSG * DD * 4);
    float* QF      = (float*)carve((size_t)BQn * DQn * 4);
    float* SF      = (float*)carve((size_t)BQn * DD * 4);
    float* Q       = (float*)carve((size_t)BQn * DQn * 4);
    float* T       = (float*)carve((size_t)BQn * 2 * DKn * 4);
    float* sb      = (float*)carve((size_t)BQn * 2 * 4);
    float* U       = (float*)carve((size_t)BQn * 2 * DKn * 4);
    float* AO      = (float*)carve((size_t)BQn * DQn * 4);
    float* O       = (float*)carve((size_t)BQn * DQn * 4);
    float* CAT     = (float*)carve((size_t)BQn * DKn * 4);
    float* MH      = (float*)carve((size_t)BQn * DD * 4);
    float* EMB     = (float*)carve((size_t)BQn * DD * 4);
    float* P1      = (float*)carve((size_t)BQn * DD * 4);
    float* P2      = (float*)carve((size_t)BQn * CDIM * 4);

    // --- memory update path -------------------------------------------------
    k_init_updidx<<<(NN_NODES + 255) / 256, 256, 0, stream>>>(upd_idx, NN_NODES);
    k_scatter_updidx<<<(MMSG + 255) / 256, 256, 0, stream>>>(upd_idx, msg_node_ids, MMSG);

    // H1 = relu(msg_raw @ msg_w1 + b1)   [8192,1024]x[1024,512]
    gemm(stream, msg_raw, 1024, 0, nullptr, msg_w1, 512, 0, 0, msg_b1, H1, 512, 0, MMSG, 512, 1024, 1);
    // Mm = H1 @ msg_w2 + b2              [8192,512]x[512,256]
    gemm(stream, H1, 512, 0, nullptr, msg_w2, 256, 0, 0, msg_b2, Mm, 256, 0, MMSG, 256, 512, 0);
    // GI = Mm @ gru_wi + bi              [8192,256]x[256,768]
    gemm(stream, Mm, 256, 0, nullptr, gru_wi, 768, 0, 0, gru_bi, GI, 768, 0, MMSG, 768, 256, 0);
    // GH = memory[ids] @ gru_wh + bh     (row gather)
    gemm(stream, memory, 256, 0, msg_node_ids, gru_wh, 768, 0, 0, gru_bh, GH, 768, 0, MMSG, 768, 256, 0);
    k_gru_combine<<<(MMSG * DD + 255) / 256, 256, 0, stream>>>(GI, GH, memory, msg_node_ids, newmem);

    // --- embedding path -----------------------------------------------------
    k_build_query<<<(BQn * DD + 255) / 256, 256, 0, stream>>>(
        node_features, memory, newmem, upd_idx, src_nodes, dst_nodes, time_b, QF, SF);

    // Q = QF @ wq + bq                   [8192,512]x[512,512]
    gemm(stream, QF, DQn, 0, nullptr, wq, DQn, 0, 0, bq, Q, DQn, 0, BQn, DQn, DQn, 0);
    k_score_bias<<<(BQn * 2 + 255) / 256, 256, 0, stream>>>(Q, bk, sb);

    // T[b,h,:] = Wk[:, h*256:(h+1)*256]^T @ q_h   (transB GEMM, no bias)
    gemm(stream, Q, DQn, 0,   nullptr, wk, DQn, 0,   1, nullptr, T, 2 * DKn, 0,   BQn, DKn, 256, 0);
    gemm(stream, Q, DQn, 256, nullptr, wk, DQn, 256, 1, nullptr, T, 2 * DKn, DKn, BQn, DKn, 256, 0);

    // fused gather + score + softmax + value-weighted key sum -> U[b,h,768]
    k_attn_fused<<<BQn, 256, 0, stream>>>(
        node_features, memory, newmem, upd_idx, edge_features, time_w, time_b,
        edge_times, neigh_ts, neigh_ids, neigh_eidx, T, sb, U);

    // AO[:, h*256:(h+1)*256] = U_h @ Wv[:, h*256:(h+1)*256] + bv_h
    gemm(stream, U, 2 * DKn, 0,   nullptr, wv, DQn, 0,   0, bv, AO, DQn, 0,   BQn, 256, DKn, 0);
    gemm(stream, U, 2 * DKn, DKn, nullptr, wv, DQn, 256, 0, bv, AO, DQn, 256, BQn, 256, DKn, 0);

    // O = AO @ wo + bo
    gemm(stream, AO, DQn, 0, nullptr, wo, DQn, 0, 0, bo, O, DQn, 0, BQn, DQn, DQn, 0);

    // merge: CAT = [O, SF]; MH = relu(CAT @ mrg_w1 + b); EMB = MH @ mrg_w2 + b
    k_build_cat<<<(BQn * DKn + 255) / 256, 256, 0, stream>>>(O, SF, CAT);
    gemm(stream, CAT, DKn, 0, nullptr, mrg_w1, DD, 0, 0, mrg_b1, MH, DD, 0, BQn, DD, DKn, 1);
    gemm(stream, MH, DD, 0, nullptr, mrg_w2, DD, 0, 0, mrg_b2, EMB, DD, 0, BQn, DD, DD, 0);

    // projector: P1 = relu(EMB @ prj_w1 + b); P2 = P1 @ prj_w2 + b; softmax
    gemm(stream, EMB, DD, 0, nullptr, prj_w1, DD, 0, 0, prj_b1, P1, DD, 0, BQn, DD, DD, 1);
    gemm(stream, P1, DD, 0, nullptr, prj_w2, CDIM, 0, 0, prj_b2, P2, CDIM, 0, BQn, CDIM, DD, 0);
    k_softmax_rows<<<(BQn + 255) / 256, 256, 0, stream>>>(P2, (float*)d_out);
}